// GraphAttentionLayer_88484916232508
// MI455X (gfx1250) — compile-verified
//
#include <hip/hip_runtime.h>
#include <hip/hip_bf16.h>
#include <math.h>

typedef __attribute__((ext_vector_type(2))) float v2f;
typedef __attribute__((ext_vector_type(8))) float v8f;

#define N_NODES   8192
#define F_DIM     256
#define THETA_C   0.5f
#define ALPHA_C   0.2f
#define NEG_INF_V (-9000000000000000.0f)

__device__ __forceinline__ v8f wmma_f32(v2f a, v2f b, v8f c) {
    // D = A(16x4 f32) * B(4x16 f32) + C(16x16 f32), wave32
    return __builtin_amdgcn_wmma_f32_16x16x4_f32(
        /*neg_a=*/false, a, /*neg_b=*/false, b,
        /*c_mod=*/(short)0, c, /*reuse_a=*/false, /*reuse_b=*/false);
}

// ---------------------------------------------------------------------------
// K1: Wh = h @ W   (8192x256 @ 256x256, fp32 WMMA 16x16x4)
// grid = (256/64, 8192/16), block = 128 (4 waves, one 16x16 tile each)
// ---------------------------------------------------------------------------
__global__ __launch_bounds__(128) void gat_wh_kernel(
    const float* __restrict__ h, const float* __restrict__ W,
    float* __restrict__ Wh) {
    const int lane    = threadIdx.x & 31;
    const int wave    = threadIdx.x >> 5;
    const int rowBase = blockIdx.y * 16;
    const int colBase = blockIdx.x * 64 + wave * 16;
    const int half    = lane >> 4;   // K sub-pair: lanes 16-31 hold K=2,3
    const int lid     = lane & 15;

    const float* ha = h + (size_t)(rowBase + lid) * F_DIM + 2 * half;
    const float* wb = W + (size_t)(2 * half) * F_DIM + colBase + lid;

    v8f acc = {};
    #pragma unroll 4
    for (int k = 0; k < F_DIM; k += 4) {
        v2f a, b;
        a.x = ha[k];
        a.y = ha[k + 1];
        b.x = wb[(size_t)k * F_DIM];
        b.y = wb[(size_t)(k + 1) * F_DIM];
        acc = wmma_f32(a, b, acc);
    }
    #pragma unroll
    for (int r = 0; r < 8; ++r) {
        int row = rowBase + r + 8 * half;   // C layout: VGPR r -> M=r / M=r+8
        Wh[(size_t)row * F_DIM + colBase + lid] = acc[r];
    }
}

// ---------------------------------------------------------------------------
// K2: f1 = Wh @ a[:256], f2 = Wh @ a[256:]. One wave per row.
// grid = 8192/8, block = 256 (8 waves)
// ---------------------------------------------------------------------------
__global__ __launch_bounds__(256) void gat_f12_kernel(
    const float* __restrict__ Wh, const float* __restrict__ a_vec,
    float* __restrict__ f1, float* __restrict__ f2) {
    const int lane = threadIdx.x & 31;
    const int wave = threadIdx.x >> 5;
    const int row  = blockIdx.x * 8 + wave;
    const float* wr = Wh + (size_t)row * F_DIM;
    float s1 = 0.f, s2 = 0.f;
    #pragma unroll
    for (int t = 0; t < 8; ++t) {
        float v = wr[lane + 32 * t];
        s1 += v * a_vec[lane + 32 * t];
        s2 += v * a_vec[F_DIM + lane + 32 * t];
    }
    #pragma unroll
    for (int off = 16; off > 0; off >>= 1) {
        s1 += __shfl_xor(s1, off, 32);
        s2 += __shfl_xor(s2, off, 32);
    }
    if (lane == 0) { f1[row] = s1; f2[row] = s2; }
}

// ---------------------------------------------------------------------------
// K3: per-row safe-softmax stats: M_i = max_j val_ij, L_i = sum_j exp(val-M)
// val_ij = adj>0 ? leakyrelu(f1_i+f2_j)+theta*cv_i : NEG_INF  (finite -9e15,
// matching the reference: an all-masked row softmaxes to uniform).
// grid = 8192 rows, block = 256 (each thread covers 32 columns)
// ---------------------------------------------------------------------------
__global__ __launch_bounds__(256) void gat_stats_kernel(
    const int* __restrict__ adj, const float* __restrict__ f1,
    const float* __restrict__ f2, const float* __restrict__ cv,
    float* __restrict__ mOut, float* __restrict__ lOut) {
    __shared__ float red[256];
    const int i   = blockIdx.x;
    const int tid = threadIdx.x;
    const float f1i = f1[i];
    const float cvi = THETA_C * cv[i];
    const int* arow = adj + (size_t)i * N_NODES;

    float vals[32];
    float m = NEG_INF_V;
    #pragma unroll
    for (int t = 0; t < 32; ++t) {
        int j = tid + 256 * t;
        float e = f1i + f2[j];
        e = (e > 0.f) ? e : ALPHA_C * e;
        e += cvi;
        float v = (arow[j] > 0) ? e : NEG_INF_V;
        vals[t] = v;
        m = fmaxf(m, v);
    }
    red[tid] = m; __syncthreads();
    for (int s = 128; s > 0; s >>= 1) {
        if (tid < s) red[tid] = fmaxf(red[tid], red[tid + s]);
        __syncthreads();
    }
    const float M = red[0];
    __syncthreads();

    float sum = 0.f;
    #pragma unroll
    for (int t = 0; t < 32; ++t) sum += __expf(vals[t] - M);
    red[tid] = sum; __syncthreads();
    for (int s = 128; s > 0; s >>= 1) {
        if (tid < s) red[tid] += red[tid + s];
        __syncthreads();
    }
    if (tid == 0) { mOut[i] = M; lOut[i] = red[0]; }
}

// ---------------------------------------------------------------------------
// K4: h' = softmax(att) @ Wh, fused with ELU.
// Attention weights are recomputed on the fly directly into the WMMA A
// fragment (16x4 f32): lane holds row = lane%16, K cols = 2*(lane>>4)+{0,1}.
// grid = 8192/16 row tiles, block = 128 (4 waves x 4 N-tiles = 256 cols).
// No barriers in the K loop.
// ---------------------------------------------------------------------------
__global__ __launch_bounds__(128) void gat_av_kernel(
    const int* __restrict__ adj, const float* __restrict__ Wh,
    const float* __restrict__ f1, const float* __restrict__ f2,
    const float* __restrict__ cv, const float* __restrict__ mRow,
    const float* __restrict__ lRow, float* __restrict__ out) {
    const int lane    = threadIdx.x & 31;
    const int wave    = threadIdx.x >> 5;
    const int rowBase = blockIdx.x * 16;
    const int colBase = wave * 64;
    const int half    = lane >> 4;
    const int lid     = lane & 15;

    const int   iA  = rowBase + lid;          // A-fragment row for this lane
    const float f1i = f1[iA];
    const float cvi = THETA_C * cv[iA];
    const float Mi  = mRow[iA];
    const float rLi = 1.0f / lRow[iA];
    const int* arow = adj + (size_t)iA * N_NODES + 2 * half;
    const float* wb = Wh + (size_t)(2 * half) * F_DIM + colBase + lid;

    v8f acc0 = {}, acc1 = {}, acc2 = {}, acc3 = {};

    for (int j = 0; j < N_NODES; j += 4) {
        __builtin_prefetch(arow + j + 512, 0, 0);   // global_prefetch_b8
        const int j0 = j + 2 * half;
        // attention probabilities -> A fragment
        float e0 = f1i + f2[j0];
        e0 = ((e0 > 0.f) ? e0 : ALPHA_C * e0) + cvi;
        float e1 = f1i + f2[j0 + 1];
        e1 = ((e1 > 0.f) ? e1 : ALPHA_C * e1) + cvi;
        float v0 = (arow[j]     > 0) ? e0 : NEG_INF_V;
        float v1 = (arow[j + 1] > 0) ? e1 : NEG_INF_V;
        v2f A;
        A.x = __expf(v0 - Mi) * rLi;
        A.y = __expf(v1 - Mi) * rLi;
        // B fragments: Wh rows j..j+3, four 16-col tiles
        const float* wbj = wb + (size_t)j * F_DIM;
        v2f B0, B1, B2, B3;
        B0.x = wbj[0];  B0.y = wbj[F_DIM];
        B1.x = wbj[16]; B1.y = wbj[F_DIM + 16];
        B2.x = wbj[32]; B2.y = wbj[F_DIM + 32];
        B3.x = wbj[48]; B3.y = wbj[F_DIM + 48];
        acc0 = wmma_f32(A, B0, acc0);
        acc1 = wmma_f32(A, B1, acc1);
        acc2 = wmma_f32(A, B2, acc2);
        acc3 = wmma_f32(A, B3, acc3);
    }

    // Epilogue: ELU(x) = x>0 ? x : exp(x)-1, store fp32
    #pragma unroll
    for (int r = 0; r < 8; ++r) {
        const int row = rowBase + r + 8 * half;
        float x0 = acc0[r], x1 = acc1[r], x2 = acc2[r], x3 = acc3[r];
        x0 = (x0 > 0.f) ? x0 : (__expf(x0) - 1.f);
        x1 = (x1 > 0.f) ? x1 : (__expf(x1) - 1.f);
        x2 = (x2 > 0.f) ? x2 : (__expf(x2) - 1.f);
        x3 = (x3 > 0.f) ? x3 : (__expf(x3) - 1.f);
        float* o = out + (size_t)row * F_DIM + colBase + lid;
        o[0]  = x0;
        o[16] = x1;
        o[32] = x2;
        o[48] = x3;
    }
}

// ---------------------------------------------------------------------------
extern "C" void kernel_launch(void* const* d_in, const int* in_sizes, int n_in,
                              void* d_out, int out_size, void* d_ws, size_t ws_size,
                              hipStream_t stream) {
    const float* h   = (const float*)d_in[0];   // (8192,256)
    const int*   adj = (const int*)  d_in[1];   // (8192,8192)
    const float* cv  = (const float*)d_in[2];   // (8192,)
    const float* W   = (const float*)d_in[3];   // (256,256)
    const float* av  = (const float*)d_in[4];   // (512,1)
    float* out = (float*)d_out;

    char* ws = (char*)d_ws;
    float* Wh = (float*)ws;                                   // 8 MB
    float* f1 = (float*)(ws + (size_t)N_NODES * F_DIM * 4);   // 32 KB
    float* f2 = f1 + N_NODES;
    float* mR = f2 + N_NODES;
    float* lR = mR + N_NODES;

    dim3 g1(F_DIM / 64, N_NODES / 16);
    gat_wh_kernel<<<g1, 128, 0, stream>>>(h, W, Wh);
    gat_f12_kernel<<<N_NODES / 8, 256, 0, stream>>>(Wh, av, f1, f2);
    gat_stats_kernel<<<N_NODES, 256, 0, stream>>>(adj, f1, f2, cv, mR, lR);
    gat_av_kernel<<<N_NODES / 16, 128, 0, stream>>>(adj, Wh, f1, f2, cv, mR, lR, out);
}